// BetaTCVAE_75204877353772
// MI455X (gfx1250) — compile-verified
//
#include <hip/hip_runtime.h>
#include <hip/hip_bf16.h>
#include <math.h>

typedef __attribute__((ext_vector_type(2))) float v2f;
typedef __attribute__((ext_vector_type(8))) float v8f;

#define BB 2048
#define LL 64
#define LOG_2PI 1.8378770664093453f
#define BETA 6.0f

// ---------------------------------------------------------------------------
// Kernel 1: per-(j,l) precompute.
//   w  = exp(-logvar), Bp[j] = [ w(64) | -2*m*w(64) ]  (K=128 row for WMMA B)
//   C[j,l] = m^2*w + logvar + log(2pi),  Csum[j] = sum_l C,  klpart[j] = row kl sum
// ---------------------------------------------------------------------------
__global__ void __launch_bounds__(64)
prep_kernel(const float* __restrict__ kl, const float* __restrict__ mean,
            const float* __restrict__ logvar,
            float* __restrict__ Bp, float* __restrict__ C,
            float* __restrict__ Csum, float* __restrict__ klpart) {
  const int j = blockIdx.x, l = threadIdx.x;
  const int idx = j * LL + l;
  const float lv = logvar[idx];
  const float mn = mean[idx];
  const float kv = kl[idx];
  const float w  = __expf(-lv);
  const float mw = mn * w;
  const float c  = fmaf(mn * mn, w, lv + LOG_2PI);
  Bp[j * 128 + l]      = w;
  Bp[j * 128 + 64 + l] = -2.0f * mw;
  C[idx] = c;

  __shared__ float sC[64], sK[64];
  sC[l] = c; sK[l] = kv;
  __syncthreads();
  for (int off = 32; off > 0; off >>= 1) {
    if (l < off) { sC[l] += sC[l + off]; sK[l] += sK[l + off]; }
    __syncthreads();
  }
  if (l == 0) { Csum[j] = sC[0]; klpart[j] = sK[0]; }
}

// ---------------------------------------------------------------------------
// Kernel 2: log_qz[i] = logsumexp_j S[i,j],  S = -0.5*(A' x B'^T + Csum[j])
// A'[i, 0:64] = z^2, A'[i, 64:128] = z.  One 16-row i-tile per block, 4 waves
// split the 128 j-tiles.  FP32 WMMA 16x16x4, K=128 => 32 chained wmma/tile.
// fp32 A layout (ISA 7.12.2): lane<16 -> {K=k0,k0+1}, lane>=16 -> {K=k0+2,k0+3}.
// C/D layout: vgpr r, lanes 0-15: M=r, N=lane; lanes 16-31: M=8+r, N=lane-16.
// ---------------------------------------------------------------------------
__global__ void __launch_bounds__(128)
logqz_kernel(const float* __restrict__ Z, const float* __restrict__ Bp,
             const float* __restrict__ Csum, float* __restrict__ logqz) {
  const int itile  = blockIdx.x;
  const int tid    = threadIdx.x;
  const int wave   = tid >> 5;
  const int lane   = tid & 31;
  const int hi     = lane >> 4;
  const int lane16 = lane & 15;

  // Hoist A fragments for the whole i-tile (invariant over j): 32 x v2f.
  v2f afrag[32];
  const int irow = itile * 16 + lane16;
#pragma unroll
  for (int s = 0; s < 32; ++s) {
    const int kb = s * 4 + hi * 2;            // even, stays within 64-half
    float x0 = Z[irow * LL + (kb & 63)];
    float x1 = Z[irow * LL + ((kb + 1) & 63)];
    if (kb < 64) { x0 *= x0; x1 *= x1; }      // first half of K = z^2
    afrag[s].x = x0; afrag[s].y = x1;
  }

  float mreg[8], sreg[8];
#pragma unroll
  for (int r = 0; r < 8; ++r) { mreg[r] = -__builtin_inff(); sreg[r] = 0.0f; }

  for (int jt = wave; jt < 128; jt += 4) {
    const int jcol = jt * 16 + lane16;        // B/C/D column owned by this lane
    const float* __restrict__ brow = Bp + (size_t)jcol * 128;
    v8f acc = {};
#pragma unroll
    for (int s = 0; s < 32; ++s) {
      const int krow = s * 4 + hi * 2;
      v2f b = *reinterpret_cast<const v2f*>(brow + krow);   // 8B-aligned
      acc = __builtin_amdgcn_wmma_f32_16x16x4_f32(
          false, afrag[s], false, b, (short)0, acc, false, false);
    }
    const float cj = Csum[jcol];
#pragma unroll
    for (int r = 0; r < 8; ++r) {             // online logsumexp over j
      const float p  = -0.5f * (acc[r] + cj);
      const float mn = fmaxf(mreg[r], p);
      sreg[r] = sreg[r] * __expf(mreg[r] - mn) + __expf(p - mn);
      mreg[r] = mn;
    }
  }

  __shared__ float lm[16 * 64], ls[16 * 64];
#pragma unroll
  for (int r = 0; r < 8; ++r) {
    const int ir  = r + hi * 8;               // local row 0..15
    const int col = lane16 + wave * 16;       // 64 partials per row
    lm[ir * 64 + col] = mreg[r];
    ls[ir * 64 + col] = sreg[r];
  }
  __syncthreads();
  if (tid < 16) {
    float M = -__builtin_inff();
    for (int c2 = 0; c2 < 64; ++c2) M = fmaxf(M, lm[tid * 64 + c2]);
    float S = 0.0f;
    for (int c2 = 0; c2 < 64; ++c2)
      S += ls[tid * 64 + c2] * __expf(lm[tid * 64 + c2] - M);
    logqz[itile * 16 + tid] = M + __logf(S);
  }
}

// ---------------------------------------------------------------------------
// Kernel 3: log_qz_product[i] = sum_l logsumexp_j p[i,j,l]  (dominant cost).
// 64 threads = all l; 8 i-rows per block (register blocking amortizes the
// L2-resident j-stream 8x).  Two-pass logsumexp: max pass, then 1 exp/elem.
// Emits tcpart[block] = sum_i (log_qz[i] - log_qz_product[i]).
// ---------------------------------------------------------------------------
__global__ void __launch_bounds__(64)
product_kernel(const float* __restrict__ Z, const float* __restrict__ Bp,
               const float* __restrict__ C, const float* __restrict__ logqz,
               float* __restrict__ tcpart) {
  const int l   = threadIdx.x;
  const int ib0 = blockIdx.x * 8;
  float zh[8], z2h[8], m[8], ss[8];
#pragma unroll
  for (int t = 0; t < 8; ++t) {
    const float z = Z[(ib0 + t) * LL + l];
    zh[t]  = -0.5f * z;            // pairs with Bp's -2*m*w => + z*m*w
    z2h[t] = -0.5f * z * z;
    m[t] = -__builtin_inff(); ss[t] = 0.0f;
  }
  for (int j = 0; j < BB; ++j) {                       // pass 1: max_j
    const float w  = Bp[j * 128 + l];
    const float w2 = Bp[j * 128 + 64 + l];
    const float ch = -0.5f * C[j * LL + l];
#pragma unroll
    for (int t = 0; t < 8; ++t) {
      const float p = fmaf(z2h[t], w, fmaf(zh[t], w2, ch));
      m[t] = fmaxf(m[t], p);
    }
  }
  for (int j = 0; j < BB; ++j) {                       // pass 2: sum exp
    const float w  = Bp[j * 128 + l];
    const float w2 = Bp[j * 128 + 64 + l];
    const float ch = -0.5f * C[j * LL + l];
#pragma unroll
    for (int t = 0; t < 8; ++t) {
      const float p = fmaf(z2h[t], w, fmaf(zh[t], w2, ch));
      ss[t] += __expf(p - m[t]);
    }
  }
  __shared__ float lds[8 * 64];
  __shared__ float diffs[8];
#pragma unroll
  for (int t = 0; t < 8; ++t) lds[t * 64 + l] = m[t] + __logf(ss[t]);
  __syncthreads();
  if (l < 8) {
    float prod = 0.0f;
    for (int c2 = 0; c2 < 64; ++c2) prod += lds[l * 64 + c2];
    diffs[l] = logqz[ib0 + l] - prod;
  }
  __syncthreads();
  if (l == 0) {
    float d = 0.0f;
    for (int t = 0; t < 8; ++t) d += diffs[t];
    tcpart[blockIdx.x] = d;
  }
}

// ---------------------------------------------------------------------------
// Kernel 4: deterministic final reduction -> scalar output.
// ---------------------------------------------------------------------------
__global__ void __launch_bounds__(256)
finalize_kernel(const float* __restrict__ klpart, const float* __restrict__ tcpart,
                float* __restrict__ out) {
  __shared__ float sk[256], st[256];
  const int tid = threadIdx.x;
  float k = 0.0f;
  for (int i = tid; i < BB; i += 256) k += klpart[i];
  sk[tid] = k;
  st[tid] = tcpart[tid];            // exactly 256 product blocks
  __syncthreads();
  for (int off = 128; off > 0; off >>= 1) {
    if (tid < off) { sk[tid] += sk[tid + off]; st[tid] += st[tid + off]; }
    __syncthreads();
  }
  if (tid == 0) out[0] = sk[0] + (BETA - 1.0f) * (st[0] / (float)BB);
}

// ---------------------------------------------------------------------------
extern "C" void kernel_launch(void* const* d_in, const int* in_sizes, int n_in,
                              void* d_out, int out_size, void* d_ws, size_t ws_size,
                              hipStream_t stream) {
  const float* kl        = (const float*)d_in[0];
  const float* z_mean    = (const float*)d_in[1];
  const float* z_logvar  = (const float*)d_in[2];
  const float* z_sampled = (const float*)d_in[3];

  float* ws     = (float*)d_ws;
  float* Bp     = ws;                 // 2048*128 = 262144
  float* C      = ws + 262144;        // 2048*64  = 131072
  float* Csum   = ws + 393216;        // 2048
  float* logqz  = ws + 395264;        // 2048
  float* klpart = ws + 397312;        // 2048
  float* tcpart = ws + 399360;        // 256   (total ~1.56 MB)

  prep_kernel<<<BB, 64, 0, stream>>>(kl, z_mean, z_logvar, Bp, C, Csum, klpart);
  logqz_kernel<<<BB / 16, 128, 0, stream>>>(z_sampled, Bp, Csum, logqz);
  product_kernel<<<BB / 8, 64, 0, stream>>>(z_sampled, Bp, C, logqz, tcpart);
  finalize_kernel<<<1, 256, 0, stream>>>(klpart, tcpart, (float*)d_out);
}